// GConvGRUModel_13864154431957
// MI455X (gfx1250) — compile-verified
//
#include <hip/hip_runtime.h>
#include <hip/hip_bf16.h>
#include <math.h>

// GConvGRU (ChebConv K=5, L=2, D=128) for MI455X / gfx1250.
// Dense GEMMs use V_WMMA_F32_16X16X4_F32 (fp32 tensor path, preserves
// reference fp32 precision), register-blocked 4 row tiles per wave so each
// B fragment feeds 4 WMMAs; A tile staged in LDS k-grouped so each A fragment
// is one aligned ds_load_b64. Sparse props via coalesced fp32 atomics.
// cheb(ones,.) exploited as rank-1 (SpMV + colsum outer product).

typedef __attribute__((ext_vector_type(2))) float v2f;
typedef __attribute__((ext_vector_type(8))) float v8f;

#define D_DIM 128
#define K_CHEB 5
#define ROWS_PB 64        // rows per block in the WMMA GEMM
#define GSTRIDE 260       // floats per kk-group in LDS: 64 rows * 4 + 4 pad

// ---------------- edge / elementwise kernels ----------------

__global__ void deg_kernel(const int* __restrict__ src, const float* __restrict__ ew,
                           float* __restrict__ deg, int E) {
  int e = blockIdx.x * blockDim.x + threadIdx.x;
  if (e < E) atomicAdd(&deg[src[e]], ew[e]);
}

__global__ void dinv_kernel(float* __restrict__ deg, int N) {
  int i = blockIdx.x * blockDim.x + threadIdx.x;
  if (i < N) { float d = deg[i]; deg[i] = d > 0.0f ? rsqrtf(d) : 0.0f; }
}

__global__ void edgew_kernel(const int* __restrict__ src, const int* __restrict__ dst,
                             const float* __restrict__ ew, const float* __restrict__ dinv,
                             float* __restrict__ w, int E) {
  int e = blockIdx.x * blockDim.x + threadIdx.x;
  if (e < E) w[e] = -dinv[src[e]] * ew[e] * dinv[dst[e]];
}

__global__ void ones_kernel(float* __restrict__ p, int n) {
  int i = blockIdx.x * blockDim.x + threadIdx.x;
  if (i < n) p[i] = 1.0f;
}

// out[dst[e]] += w[e] * tin[src[e]]   (scalar SpMV for the rank-1 ones path)
__global__ void spmv_kernel(const int* __restrict__ src, const int* __restrict__ dst,
                            const float* __restrict__ w, const float* __restrict__ tin,
                            float* __restrict__ tout, int E) {
  int e = blockIdx.x * blockDim.x + threadIdx.x;
  if (e < E) atomicAdd(&tout[dst[e]], w[e] * tin[src[e]]);
}

// a = 2*a - b   (Chebyshev recurrence combine)
__global__ void combine_kernel(float* __restrict__ a, const float* __restrict__ b, int n) {
  int i = blockIdx.x * blockDim.x + threadIdx.x;
  if (i < n) a[i] = 2.0f * a[i] - b[i];
}

// out[dst[e]*128 + d] += w[e] * x[src[e]*128 + d]   (SpMM via atomics)
__global__ void spmm_kernel(const int* __restrict__ src, const int* __restrict__ dst,
                            const float* __restrict__ w, const float* __restrict__ x,
                            float* __restrict__ out, int E) {
  int idx = blockIdx.x * blockDim.x + threadIdx.x;   // E*128 threads
  int e = idx >> 7;
  int d = idx & 127;
  if (e < E) {
    int s = src[e], t = dst[e];
    atomicAdd(&out[t * D_DIM + d], w[e] * x[s * D_DIM + d]);
  }
}

// csum[g*K+k][j] = sum_i Wh_l[g][k][i][j]   for g in {0,1}
__global__ void csum_kernel(const float* __restrict__ Wh_l, float* __restrict__ csum) {
  int idx = blockIdx.x * blockDim.x + threadIdx.x;   // 2*K*128 threads
  if (idx < 2 * K_CHEB * D_DIM) {
    int j = idx & 127;
    int gk = idx >> 7;
    const float* Wp = Wh_l + (size_t)gk * D_DIM * D_DIM;
    float s = 0.0f;
    for (int i = 0; i < D_DIM; ++i) s += Wp[i * D_DIM + j];
    csum[idx] = s;
  }
}

// Z = sigmoid(Gz + rank1(t,csumZ) + bh0), R = sigmoid(Gr + rank1(t,csumR) + bh1); in-place.
__global__ void gate1_kernel(float* __restrict__ Gz, float* __restrict__ Gr,
                             const float* __restrict__ tones,  // [K][N]
                             const float* __restrict__ csum,   // [2][K][128]
                             const float* __restrict__ bh0, const float* __restrict__ bh1,
                             int N) {
  int idx = blockIdx.x * blockDim.x + threadIdx.x;
  if (idx < N * D_DIM) {
    int n = idx >> 7;
    int j = idx & 127;
    float tz = 0.0f, tr = 0.0f;
#pragma unroll
    for (int k = 0; k < K_CHEB; ++k) {
      float t = tones[(size_t)k * N + n];
      tz += t * csum[k * D_DIM + j];
      tr += t * csum[(K_CHEB + k) * D_DIM + j];
    }
    float z = Gz[idx] + tz + bh0[j];
    float r = Gr[idx] + tr + bh1[j];
    Gz[idx] = 1.0f / (1.0f + expf(-z));
    Gr[idx] = 1.0f / (1.0f + expf(-r));
  }
}

// h = relu(Z + (1-Z)*tanh(Gh + bh2))
__global__ void gate2_kernel(const float* __restrict__ Gz, const float* __restrict__ Gh,
                             const float* __restrict__ bh2, float* __restrict__ h, int N) {
  int idx = blockIdx.x * blockDim.x + threadIdx.x;
  if (idx < N * D_DIM) {
    int j = idx & 127;
    float z = Gz[idx];
    float ht = tanhf(Gh[idx] + bh2[j]);
    float hn = z + (1.0f - z) * ht;
    h[idx] = hn > 0.0f ? hn : 0.0f;
  }
}

// ---------------- WMMA fp32 GEMM core ----------------
// Block = 256 threads = 8 waves covering a 64x128 output strip; wave w owns
// columns [16w,16w+16) and 4 row tiles (each B fragment feeds 4 WMMAs).
// LDS A layout: for kk-group g (4 consecutive k), Ast[g*GSTRIDE + 4*m + (k&3)].
// A fragment (k0,k0+1), k0 = 4g + 2*hi, is one aligned 8-byte LDS load.
// Reads: lanes 0-15 hit banks 4(g+m)+{0,1}, lanes 16-31 the complementary
// +{2,3} set -> conflict-free.
// A frag layout (V_WMMA_F32_16X16X4_F32): lane<16 -> M=lane, K={k0,k0+1}; lane>=16 -> K={k0+2,k0+3}.
// C/D layout: vgpr r -> row = 8*(lane>=16) + r, col = lane&15.

__device__ __forceinline__ void stage_A(const float* __restrict__ A, float* __restrict__ Ast,
                                        int rowBase, int tid, int N) {
#pragma unroll
  for (int i = 0; i < 8; ++i) {
    int idx = tid + i * 256;          // 0..2047 float4 slots
    int mm = idx >> 5;                // row 0..63
    int q  = idx & 31;                // kk-group 0..31
    int row = rowBase + mm;
    float4 v = make_float4(0.0f, 0.0f, 0.0f, 0.0f);
    if (row < N) v = *(const float4*)&A[(size_t)row * D_DIM + 4 * q];
    *(float4*)&Ast[q * GSTRIDE + 4 * mm] = v;
  }
}

__device__ __forceinline__ void gemm_mma_pass(
    const float* __restrict__ Ast, const float* __restrict__ W,
    const float* __restrict__ bias, float* __restrict__ Cout,
    int accumulate, int N, int rowBase, int m, int hi, int bcol) {
  v8f acc0 = {}, acc1 = {}, acc2 = {}, acc3 = {};
#pragma unroll
  for (int kk = 0; kk < D_DIM; kk += 4) {
    int g = kk >> 2;
    int k0 = kk + 2 * hi;
    v2f b;
    b.x = W[(size_t)k0 * D_DIM + bcol];
    b.y = W[(size_t)(k0 + 1) * D_DIM + bcol];
    const float* base = Ast + g * GSTRIDE + 2 * hi + 4 * m;
    v2f a0 = *(const v2f*)(base);
    v2f a1 = *(const v2f*)(base + 64);
    v2f a2 = *(const v2f*)(base + 128);
    v2f a3 = *(const v2f*)(base + 192);
    acc0 = __builtin_amdgcn_wmma_f32_16x16x4_f32(false, a0, false, b, (short)0, acc0, false, false);
    acc1 = __builtin_amdgcn_wmma_f32_16x16x4_f32(false, a1, false, b, (short)0, acc1, false, false);
    acc2 = __builtin_amdgcn_wmma_f32_16x16x4_f32(false, a2, false, b, (short)0, acc2, false, false);
    acc3 = __builtin_amdgcn_wmma_f32_16x16x4_f32(false, a3, false, b, (short)0, acc3, false, false);
  }
  float bval = bias ? bias[bcol] : 0.0f;
  const v8f* accs[4] = {&acc0, &acc1, &acc2, &acc3};
#pragma unroll
  for (int t = 0; t < 4; ++t) {
    const v8f& acc = *accs[t];
#pragma unroll
    for (int r = 0; r < 8; ++r) {
      int row = rowBase + t * 16 + hi * 8 + r;
      if (row < N) {
        size_t off = (size_t)row * D_DIM + bcol;
        float v = acc[r] + bval;
        if (accumulate) v += Cout[off];
        Cout[off] = v;
      }
    }
  }
}

// Single-output GEMM: C (+)= A @ W (+ bias)
__global__ __launch_bounds__(256) void cheb_gemm64(
    const float* __restrict__ A, const float* __restrict__ W,
    const float* __restrict__ bias, float* __restrict__ Cout,
    int accumulate, int N) {
  __shared__ float Ast[32 * GSTRIDE];
  int tid = threadIdx.x;
  int rowBase = blockIdx.x * ROWS_PB;
  stage_A(A, Ast, rowBase, tid, N);
  __syncthreads();
  int lane = tid & 31;
  int m = lane & 15, hi = lane >> 4;
  int bcol = (tid >> 5) * 16 + m;
  gemm_mma_pass(Ast, W, bias, Cout, accumulate, N, rowBase, m, hi, bcol);
}

// Triple-output GEMM (shared A tile): Cg (+)= A @ Wg (+ bg), g = 0..2
__global__ __launch_bounds__(256) void cheb_gemm64x3(
    const float* __restrict__ A,
    const float* __restrict__ W0, const float* __restrict__ W1, const float* __restrict__ W2,
    const float* __restrict__ b0, const float* __restrict__ b1, const float* __restrict__ b2,
    float* __restrict__ C0, float* __restrict__ C1, float* __restrict__ C2,
    int accumulate, int N) {
  __shared__ float Ast[32 * GSTRIDE];
  int tid = threadIdx.x;
  int rowBase = blockIdx.x * ROWS_PB;
  stage_A(A, Ast, rowBase, tid, N);
  __syncthreads();
  int lane = tid & 31;
  int m = lane & 15, hi = lane >> 4;
  int bcol = (tid >> 5) * 16 + m;
  gemm_mma_pass(Ast, W0, b0, C0, accumulate, N, rowBase, m, hi, bcol);
  gemm_mma_pass(Ast, W1, b1, C1, accumulate, N, rowBase, m, hi, bcol);
  gemm_mma_pass(Ast, W2, b2, C2, accumulate, N, rowBase, m, hi, bcol);
}

// ---------------- host orchestration ----------------

static inline dim3 g1(int n) { return dim3((unsigned)((n + 255) / 256)); }

extern "C" void kernel_launch(void* const* d_in, const int* in_sizes, int n_in,
                              void* d_out, int out_size, void* d_ws, size_t ws_size,
                              hipStream_t stream) {
  const int*   edge_index = (const int*)d_in[0];    // [2,E]
  const float* ew   = (const float*)d_in[1];        // [E]
  const float* emb  = (const float*)d_in[2];        // [N,128]
  const float* Wx   = (const float*)d_in[3];        // [L,3,K,128,128]
  const float* bx   = (const float*)d_in[4];        // [L,3,128]
  const float* Wh   = (const float*)d_in[5];
  const float* bh   = (const float*)d_in[6];

  const int E = in_sizes[1];
  const int N = in_sizes[2] / D_DIM;
  const int L = in_sizes[3] / (3 * K_CHEB * D_DIM * D_DIM);
  const int* src = edge_index;
  const int* dst = edge_index + E;

  const size_t ND = (size_t)N * D_DIM;
  const size_t DD = (size_t)D_DIM * D_DIM;

  // workspace (floats): 3 T + 3 G (N*128 each) + w[E] + deg[N] + tones[K*N] + csum[2*K*128]
  float* ws = (float*)d_ws;
  float* T0 = ws;
  float* T1 = ws + ND;
  float* T2 = ws + 2 * ND;
  float* Gz = ws + 3 * ND;
  float* Gr = ws + 4 * ND;
  float* Gh = ws + 5 * ND;
  float* wbuf  = ws + 6 * ND;
  float* deg   = wbuf + E;                 // becomes dinv in-place
  float* tones = deg + N;                  // [K][N]
  float* csum  = tones + (size_t)K_CHEB * N;
  float* Tbuf[3] = {T0, T1, T2};
  float* hbuf = (float*)d_out;             // layer hidden state lives in d_out

  // ---- graph preprocessing (normalized Laplacian edge weights) ----
  hipMemsetAsync(deg, 0, (size_t)N * sizeof(float), stream);
  deg_kernel<<<g1(E), 256, 0, stream>>>(src, ew, deg, E);
  dinv_kernel<<<g1(N), 256, 0, stream>>>(deg, N);
  edgew_kernel<<<g1(E), 256, 0, stream>>>(src, dst, ew, deg, wbuf, E);

  // ---- t_k = T_k(L_hat) @ ones_N  (rank-1 basis for cheb(ones)) ----
  ones_kernel<<<g1(N), 256, 0, stream>>>(tones, N);
  for (int k = 1; k < K_CHEB; ++k) {
    float* tn = tones + (size_t)k * N;
    hipMemsetAsync(tn, 0, (size_t)N * sizeof(float), stream);
    spmv_kernel<<<g1(E), 256, 0, stream>>>(src, dst, wbuf, tones + (size_t)(k - 1) * N, tn, E);
    if (k >= 2)
      combine_kernel<<<g1(N), 256, 0, stream>>>(tn, tones + (size_t)(k - 2) * N, N);
  }

  const float* x = emb;
  const int gemmGrid = (N + ROWS_PB - 1) / ROWS_PB;
  const int spmmN = E * D_DIM;

  for (int l = 0; l < L; ++l) {
    const float* Wx_l = Wx + (size_t)l * 3 * K_CHEB * DD;
    const float* Wh_l = Wh + (size_t)l * 3 * K_CHEB * DD;
    const float* bx_l = bx + (size_t)l * 3 * D_DIM;
    const float* bh_l = bh + (size_t)l * 3 * D_DIM;

    csum_kernel<<<g1(2 * K_CHEB * D_DIM), 256, 0, stream>>>(Wh_l, csum);

    // ----- Chebyshev basis of x, streamed into the three gate GEMMs -----
    hipMemcpyAsync(Tbuf[0], x, ND * sizeof(float), hipMemcpyDeviceToDevice, stream);
    cheb_gemm64x3<<<gemmGrid, 256, 0, stream>>>(
        Tbuf[0],
        Wx_l, Wx_l + (size_t)K_CHEB * DD, Wx_l + (size_t)2 * K_CHEB * DD,
        bx_l, bx_l + D_DIM, bx_l + 2 * D_DIM,
        Gz, Gr, Gh, 0, N);
    for (int k = 1; k < K_CHEB; ++k) {
      float* Tn  = Tbuf[k % 3];
      float* Tp  = Tbuf[(k - 1) % 3];
      float* Tpp = Tbuf[(k + 1) % 3];      // == (k-2) mod 3
      hipMemsetAsync(Tn, 0, ND * sizeof(float), stream);
      spmm_kernel<<<g1(spmmN), 256, 0, stream>>>(src, dst, wbuf, Tp, Tn, E);
      if (k >= 2)
        combine_kernel<<<g1((int)ND), 256, 0, stream>>>(Tn, Tpp, (int)ND);
      cheb_gemm64x3<<<gemmGrid, 256, 0, stream>>>(
          Tn,
          Wx_l + (size_t)k * DD,
          Wx_l + ((size_t)K_CHEB + k) * DD,
          Wx_l + ((size_t)2 * K_CHEB + k) * DD,
          nullptr, nullptr, nullptr,
          Gz, Gr, Gh, 1, N);
    }

    // Z, R (adds rank-1 cheb(ones,Wh0/Wh1) + biases), in-place in Gz/Gr
    gate1_kernel<<<g1((int)ND), 256, 0, stream>>>(Gz, Gr, tones, csum,
                                                  bh_l, bh_l + D_DIM, N);

    // ----- Chebyshev basis of R, accumulated into Gh with Wh[l][2] -----
    hipMemcpyAsync(Tbuf[0], Gr, ND * sizeof(float), hipMemcpyDeviceToDevice, stream);
    cheb_gemm64<<<gemmGrid, 256, 0, stream>>>(Tbuf[0], Wh_l + (size_t)2 * K_CHEB * DD,
                                              nullptr, Gh, 1, N);
    for (int k = 1; k < K_CHEB; ++k) {
      float* Tn  = Tbuf[k % 3];
      float* Tp  = Tbuf[(k - 1) % 3];
      float* Tpp = Tbuf[(k + 1) % 3];
      hipMemsetAsync(Tn, 0, ND * sizeof(float), stream);
      spmm_kernel<<<g1(spmmN), 256, 0, stream>>>(src, dst, wbuf, Tp, Tn, E);
      if (k >= 2)
        combine_kernel<<<g1((int)ND), 256, 0, stream>>>(Tn, Tpp, (int)ND);
      cheb_gemm64<<<gemmGrid, 256, 0, stream>>>(Tn, Wh_l + ((size_t)2 * K_CHEB + k) * DD,
                                                nullptr, Gh, 1, N);
    }

    // h = relu(Z + (1-Z)*tanh(Gh + bh2)) -> d_out (next layer's x)
    gate2_kernel<<<g1((int)ND), 256, 0, stream>>>(Gz, Gh, bh_l + 2 * D_DIM, hbuf, N);
    x = hbuf;
  }
}